// ChamferDistanceL2_19705309954521
// MI455X (gfx1250) — compile-verified
//
#include <hip/hip_runtime.h>
#include <hip/hip_bf16.h>

#define B_   16
#define N_   4096
#define M_   4096
#define WAVES 8
#define CHUNK 512            // target rows staged in LDS per iteration

typedef __attribute__((ext_vector_type(2))) float v2f;
typedef __attribute__((ext_vector_type(4))) float v4f;
typedef __attribute__((ext_vector_type(8))) float v8f;

// Order-preserving float<->int key (handles negatives); self-inverse.
__device__ __forceinline__ int f2key(float f) {
    int i = __float_as_int(f);
    return i >= 0 ? i : (i ^ 0x7fffffff);
}
__device__ __forceinline__ float key2f(int k) {
    int i = k >= 0 ? k : (k ^ 0x7fffffff);
    return __int_as_float(i);
}

// ---------------------------------------------------------------------------
// Kernel 1: init per-m min keys to +inf
// ---------------------------------------------------------------------------
__global__ void chamfer_init(int* __restrict__ dst_key, int n) {
    int i = blockIdx.x * blockDim.x + threadIdx.x;
    if (i < n) dst_key[i] = 0x7F800000;  // key(+inf) == bits of +inf
}

// ---------------------------------------------------------------------------
// Kernel 2: WMMA distance tiles + min reductions.
// Block = 256 threads = 8 waves; each wave owns a 16-row n-tile.
// dist[n,m] = s2[n] + d2[m] - 2 p.t  via ONE v_wmma_f32_16x16x4_f32 per tile:
//   A[n,:] = (-2p0,-2p1,-2p2, 1), B[:,m] = (t0,t1,t2,d2[m]), C = s2[n] rows.
// Inner loop is branch-free: b64 B-frag load (half folded into the address),
// software-pipelined one tile ahead; per-half column mins go straight to LDS.
// ---------------------------------------------------------------------------
__global__ __launch_bounds__(256) void chamfer_main(const float* __restrict__ pred,
                                                    const float* __restrict__ target,
                                                    float* __restrict__ src_min,
                                                    int*   __restrict__ dst_key) {
    __shared__ float tgt[(CHUNK + 16) * 4];       // +1 tile pad for pipelined preload
    __shared__ float minbuf[WAVES * 2 * CHUNK];   // [wave][m][half], halves interleaved

    const int tid  = threadIdx.x;
    const int wv   = tid >> 5;
    const int lane = tid & 31;
    const int half = lane >> 4;               // 0: K={0,1}, 1: K={2,3}
    const int l16  = lane & 15;
    const int b    = blockIdx.y;
    const int n0   = blockIdx.x * (WAVES * 16) + wv * 16;

    // ---- A fragment + C (s2 per row), fixed for the whole m loop ----
    const int   row = n0 + l16;
    const float* p  = pred + ((size_t)b * N_ + row) * 3;
    const float p0 = p[0], p1 = p[1], p2 = p[2];
    const float s2 = p0 * p0 + p1 * p1 + p2 * p2;   // lane l holds s2[n0 + l%16]

    v2f afrag;
    afrag.x = half ? (-2.0f * p2) : (-2.0f * p0);
    afrag.y = half ? 1.0f         : (-2.0f * p1);

    v8f cinit;
#pragma unroll
    for (int r = 0; r < 8; ++r)
        cinit[r] = __shfl(s2, r + half * 8, 32);    // row n0 + r + 8*half

    float racc[8];
#pragma unroll
    for (int r = 0; r < 8; ++r) racc[r] = 3.0e38f;

    const float* tbase = target + (size_t)b * M_ * 3;
    // Per-lane LDS slots (half baked into the B-frag read address).
    const float* bslot   = &tgt[l16 * 4 + half * 2];
    float*       mslot   = &minbuf[wv * (2 * CHUNK) + l16 * 2 + half];

    for (int ch = 0; ch < M_; ch += CHUNK) {
        // ---- stage target chunk into LDS (cooperative, 2 rows/thread) ----
        for (int i = tid; i < CHUNK; i += 256) {
            const float* t = tbase + (size_t)(ch + i) * 3;
            float t0 = t[0], t1 = t[1], t2 = t[2];
            v4f tv; tv.x = t0; tv.y = t1; tv.z = t2; tv.w = t0*t0 + t1*t1 + t2*t2;
            *(v4f*)(&tgt[i * 4]) = tv;
        }
        if (ch + CHUNK < M_)                   // hint next chunk toward cache
            __builtin_prefetch(tbase + (size_t)(ch + CHUNK) * 3 + tid * 24, 0, 1);
        __syncthreads();

        // ---- 32 WMMA tiles per chunk, B-frag pipelined one tile ahead ----
        v2f bf = *(const v2f*)(bslot);
        for (int tile = 0; tile < CHUNK; tile += 16) {
            v2f cur = bf;
            bf = *(const v2f*)(bslot + (tile + 16) * 4);   // pad keeps this in-bounds

            v8f d = __builtin_amdgcn_wmma_f32_16x16x4_f32(
                false, afrag, false, cur, (short)0, cinit, false, false);

            // per-n accumulation (min over m) + per-half column min of this tile
            float cmin = fminf(d[0], d[1]);
#pragma unroll
            for (int r = 2; r < 8; ++r) cmin = fminf(cmin, d[r]);
#pragma unroll
            for (int r = 0; r < 8; ++r) racc[r] = fminf(racc[r], d[r]);

            mslot[tile * 2] = cmin;            // unpredicated, conflict-free
        }
        __syncthreads();

        // ---- combine 8 waves x 2 halves, one global atomic-min per m ----
        for (int i = tid; i < CHUNK; i += 256) {
            float v = 3.0e38f;
#pragma unroll
            for (int w2 = 0; w2 < WAVES; ++w2) {
                v2f pr = *(const v2f*)(&minbuf[w2 * (2 * CHUNK) + i * 2]);
                v = fminf(v, fminf(pr.x, pr.y));
            }
            atomicMin(&dst_key[b * M_ + ch + i], f2key(v));
        }
        __syncthreads();
    }

    // ---- finalize per-n mins: tree-min across the 16-lane halves ----
#pragma unroll
    for (int r = 0; r < 8; ++r) {
        float v = racc[r];
        v = fminf(v, __shfl_xor(v, 1, 32));
        v = fminf(v, __shfl_xor(v, 2, 32));
        v = fminf(v, __shfl_xor(v, 4, 32));
        v = fminf(v, __shfl_xor(v, 8, 32));
        racc[r] = v;
    }
    if (l16 == 0) {   // lane 0 -> rows n0..n0+7 ; lane 16 -> rows n0+8..n0+15
#pragma unroll
        for (int r = 0; r < 8; ++r)
            src_min[(size_t)b * N_ + n0 + half * 8 + r] = racc[r];
    }
}

// ---------------------------------------------------------------------------
// Kernel 3: final reduction to scalar loss
// loss = mean_b( mean_n src_min + mean_m dst_min )
// ---------------------------------------------------------------------------
__global__ __launch_bounds__(256) void chamfer_reduce(const float* __restrict__ src_min,
                                                      const int*   __restrict__ dst_key,
                                                      float* __restrict__ out) {
    __shared__ float sbuf[256];
    const int tid = threadIdx.x;

    float a1 = 0.0f, a2 = 0.0f;
    for (int i = tid; i < B_ * N_; i += 256) a1 += src_min[i];
    for (int i = tid; i < B_ * M_; i += 256) a2 += key2f(dst_key[i]);

    sbuf[tid] = a1;
    __syncthreads();
    for (int s = 128; s > 0; s >>= 1) {
        if (tid < s) sbuf[tid] += sbuf[tid + s];
        __syncthreads();
    }
    float sum1 = sbuf[0];
    __syncthreads();

    sbuf[tid] = a2;
    __syncthreads();
    for (int s = 128; s > 0; s >>= 1) {
        if (tid < s) sbuf[tid] += sbuf[tid + s];
        __syncthreads();
    }
    if (tid == 0)
        out[0] = sum1 * (1.0f / ((float)B_ * (float)N_)) +
                 sbuf[0] * (1.0f / ((float)B_ * (float)M_));
}

// ---------------------------------------------------------------------------
extern "C" void kernel_launch(void* const* d_in, const int* in_sizes, int n_in,
                              void* d_out, int out_size, void* d_ws, size_t ws_size,
                              hipStream_t stream) {
    (void)in_sizes; (void)n_in; (void)out_size; (void)ws_size;
    const float* pred   = (const float*)d_in[0];   // (B, N, 3) f32
    const float* target = (const float*)d_in[1];   // (B, M, 3) f32
    float* out = (float*)d_out;                    // scalar f32

    // ws layout: [B*N floats src->dst mins][B*M ints dst->src min keys]
    float* ws_src = (float*)d_ws;
    int*   ws_dst = (int*)((char*)d_ws + (size_t)B_ * N_ * sizeof(float));

    const int nkeys = B_ * M_;
    chamfer_init<<<(nkeys + 255) / 256, 256, 0, stream>>>(ws_dst, nkeys);

    dim3 grid(N_ / (WAVES * 16), B_);   // (32, 16)
    chamfer_main<<<grid, 256, 0, stream>>>(pred, target, ws_src, ws_dst);

    chamfer_reduce<<<1, 256, 0, stream>>>(ws_src, ws_dst, out);
}